// KLTStereoSRQA_31954556682566
// MI455X (gfx1250) — compile-verified
//
#include <hip/hip_runtime.h>
#include <math.h>

#define HH 512
#define WW 512
#define NPATCH 4096
#define NFEAT 585
#define NGAM 9801
#define NB 8

typedef __attribute__((ext_vector_type(2))) float v2f;
typedef __attribute__((ext_vector_type(8))) float v8f;

__device__ const float CM[3][3] = {
    {0.06f, 0.63f, 0.27f},
    {0.30f, 0.04f, -0.35f},
    {0.34f, -0.60f, 0.17f}};

__device__ __forceinline__ int refl(int q) {
    if (q < 0) return -q;
    if (q >= HH) return 2 * HH - 2 - q;
    return q;
}

// ---------------------------------------------------------------- Γ table
__global__ void gamma_table_kernel(float* __restrict__ gam, float* __restrict__ rgam) {
    int i = blockIdx.x * blockDim.x + threadIdx.x;
    if (i >= NGAM) return;
    float g = 0.2f + 0.001f * (float)i;
    gam[i] = g;
    float lg = 2.0f * lgammaf(2.0f / g) - lgammaf(1.0f / g) - lgammaf(3.0f / g);
    rgam[i] = expf(lg);
}

// ------------------------------------------- KLT kernel repack (once/launch)
// kpack[c][i][j] = kern[(j*64 + i)*3 + c]  (A = K_c^T laid out row-contiguous
// in j so WMMA A-fragment pairs become single b64 loads with imm offsets)
__global__ void repack_kernel(const float* __restrict__ kern, float* __restrict__ kpack) {
    int t = blockIdx.x * blockDim.x + threadIdx.x;
    if (t >= 64 * 64 * 3) return;
    int j = t & 63;
    int i = (t >> 6) & 63;
    int c = t >> 12;
    kpack[((size_t)(c * 64 + i) << 6) + j] = kern[((size_t)j * 64 + i) * 3 + c];
}

// --------------------------------------------- fused color+blur+MSCN kernel
// One block = one 32x32 output tile of one (n, oc) plane.
// Loads 38x38 halo of color-transformed values into LDS, does separable
// 7x7 blur of o and o^2 in LDS, emits mscn = (o - mu) / (sigma + 1).
__global__ __launch_bounds__(256) void mscn_kernel(const float* __restrict__ img,
                                                   int ch_off,
                                                   float* __restrict__ out) {
    __shared__ float so[38 * 38];
    __shared__ float ho[38 * 32];
    __shared__ float ho2[38 * 32];

    int bid = blockIdx.x;
    int tile = bid & 255;
    int plane = bid >> 8;            // n*3 + oc
    int n = plane / 3, oc = plane % 3;
    int ty = (tile >> 4) * 32, tx = (tile & 15) * 32;
    int tid = threadIdx.x;

    float wgt[7];
    {
        float s = 0.f;
        #pragma unroll
        for (int i = 0; i < 7; ++i) {
            float t = (float)(i - 3) * (6.0f / 7.0f);   // t/sigma, sigma=7/6
            wgt[i] = expf(-0.5f * t * t);
            s += wgt[i];
        }
        #pragma unroll
        for (int i = 0; i < 7; ++i) wgt[i] /= s;
    }

    const float m0 = CM[oc][0], m1 = CM[oc][1], m2 = CM[oc][2];
    const float* p0 = img + (size_t)(n * 6 + ch_off + 0) * HH * WW;
    const float* p1 = img + (size_t)(n * 6 + ch_off + 1) * HH * WW;
    const float* p2 = img + (size_t)(n * 6 + ch_off + 2) * HH * WW;

    for (int idx = tid; idx < 38 * 38; idx += 256) {
        int ly = idx / 38, lx = idx - ly * 38;
        int gy = refl(ty - 3 + ly), gx = refl(tx - 3 + lx);
        int g = gy * WW + gx;
        so[idx] = m0 * p0[g] + m1 * p1[g] + m2 * p2[g];
    }
    __syncthreads();

    for (int idx = tid; idx < 38 * 32; idx += 256) {
        int r = idx >> 5, c = idx & 31;
        const float* row = &so[r * 38 + c];
        float a = 0.f, a2 = 0.f;
        #pragma unroll
        for (int i = 0; i < 7; ++i) {
            float v = row[i];
            a += wgt[i] * v;
            a2 += wgt[i] * v * v;
        }
        ho[idx] = a;
        ho2[idx] = a2;
    }
    __syncthreads();

    float* outp = out + ((size_t)plane * HH + ty) * WW + tx;
    for (int idx = tid; idx < 32 * 32; idx += 256) {
        int y = idx >> 5, x = idx & 31;
        float mu = 0.f, e2 = 0.f;
        #pragma unroll
        for (int i = 0; i < 7; ++i) {
            mu += wgt[i] * ho[(y + i) * 32 + x];
            e2 += wgt[i] * ho2[(y + i) * 32 + x];
        }
        float ov = so[(y + 3) * 38 + (x + 3)];
        float sg = sqrtf(fabsf(e2 - mu * mu));
        outp[(size_t)y * WW + x] = (ov - mu) / (sg + 1.0f);
    }
}

// ------------------------------------------------- KLT GEMM via f32 WMMA
// klt[n,c,i,p] = sum_j kpack[c,i,j] * pm[n,c,j,p],  M=64, N=4096, K=64.
// One wave = one (n,c, 16-wide p tile): 4 i-tiles x 16 K-steps of
// v_wmma_f32_16x16x4_f32. B fragments loaded once (b64-vectorized), reused
// across i-tiles; A fragments are b64 loads with immediate offsets.
__global__ __launch_bounds__(256) void klt_wmma_kernel(const float* __restrict__ mscn,
                                                       const float* __restrict__ kpack,
                                                       float* __restrict__ klt) {
    int wid = blockIdx.x * 8 + (threadIdx.x >> 5);
    int lane = threadIdx.x & 31;
    int pt = wid & 255;
    int c = (wid >> 8) % 3;
    int n = wid / 768;
    int p_base = pt * 16;
    int m = p_base >> 6;          // patch row (constant within 16-wide tile)
    int nn0 = p_base & 63;        // first patch column
    int lp = lane & 15;
    int hi = lane >> 4;           // lane half selects K pair / M half

    // B fragment source: pm[j, p] = mscn[n,c, m*8 + j/8, (nn0+lp)*8 + j%8]
    // For lane half hi, row r contributes cc in {2hi,2hi+1, 2hi+4,2hi+5}:
    //   j = 8r + 2hi + d      -> kk = 2r,   v = d
    //   j = 8r + 4 + 2hi + d  -> kk = 2r+1, v = d
    const float* src = mscn + ((size_t)(n * 3 + c) * HH + m * 8) * WW + (nn0 + lp) * 8;
    float bf[32];
    #pragma unroll
    for (int r = 0; r < 8; ++r) {
        const float* rowp = src + r * WW + hi * 2;
        v2f q0 = *(const v2f*)(rowp);
        v2f q1 = *(const v2f*)(rowp + 4);
        bf[(2 * r) * 2 + 0] = q0.x;
        bf[(2 * r) * 2 + 1] = q0.y;
        bf[(2 * r + 1) * 2 + 0] = q1.x;
        bf[(2 * r + 1) * 2 + 1] = q1.y;
    }

    float* dst = klt + (size_t)(n * 3 + c) * 64 * NPATCH + p_base;

    for (int it = 0; it < 4; ++it) {
        int ib = it * 16;
        // A row for this lane: kpack[c][ib+lp][j], j0 = kk*4 + hi*2
        const float* abase = kpack + ((size_t)(c * 64 + ib + lp) << 6) + hi * 2;
        v8f acc = {};
        #pragma unroll
        for (int kk = 0; kk < 16; ++kk) {
            v2f a = *(const v2f*)(abase + kk * 4);
            v2f b;
            b.x = bf[kk * 2 + 0];
            b.y = bf[kk * 2 + 1];
            acc = __builtin_amdgcn_wmma_f32_16x16x4_f32(
                false, a, false, b, (short)0, acc, false, false);
        }
        #pragma unroll
        for (int v = 0; v < 8; ++v) {
            int Mi = hi * 8 + v;                       // C/D layout
            dst[(size_t)(ib + Mi) * NPATCH + lp] = acc[v];
        }
    }
}

// ---------------------------------------------- per-row AGGD reductions
// stats[row][0..5] = cnt_neg, cnt_pos, sum_neg_sq, sum_pos_sq, sum_abs, sum_sq
__global__ __launch_bounds__(256) void rowstats_kernel(const float* __restrict__ klt,
                                                       float* __restrict__ stats) {
    __shared__ float sd[256];
    int row = blockIdx.x;                 // (n*3+c)*64 + i
    const float* p = klt + (size_t)row * NPATCH;
    int tid = threadIdx.x;
    float cn = 0, cp = 0, sn = 0, sp = 0, sa = 0, ss = 0;
    for (int i = tid; i < NPATCH; i += 256) {
        float v = p[i];
        float v2 = v * v;
        if (v < 0.f)      { cn += 1.f; sn += v2; }
        else if (v > 0.f) { cp += 1.f; sp += v2; }
        sa += fabsf(v);
        ss += v2;
    }
    float vals[6] = {cn, cp, sn, sp, sa, ss};
    for (int s = 0; s < 6; ++s) {
        sd[tid] = vals[s];
        __syncthreads();
        for (int off = 128; off > 0; off >>= 1) {
            if (tid < off) sd[tid] += sd[tid + off];
            __syncthreads();
        }
        if (tid == 0) stats[(size_t)row * 6 + s] = sd[0];
        __syncthreads();
    }
}

__device__ __forceinline__ void aggd_finish(float nneg, float npos, float sn, float sp,
                                            float sa, float ss, float Minv,
                                            const float* __restrict__ gam,
                                            const float* __restrict__ rgam,
                                            float* out3) {
    float leftstd = sqrtf(sn / nneg);
    float rightstd = sqrtf(sp / npos);
    float g = leftstd / rightstd;
    float ma = sa * Minv;
    float ms = ss * Minv;
    float rhat = (ma * ma) / ms;
    float g2 = g * g;
    float rhatnorm = rhat * (g2 * g + 1.0f) * (g + 1.0f) / ((g2 + 1.0f) * (g2 + 1.0f));
    int best = 0;
    float bd = 3.4e38f;
    for (int j = 0; j < NGAM; ++j) {
        float d = rgam[j] - rhatnorm;
        d = d * d;
        if (d < bd) { bd = d; best = j; }   // first minimum, like jnp.argmin
    }
    out3[0] = gam[best];
    out3[1] = leftstd;
    out3[2] = rightstd;
}

// per-(i,c) AGGD: counts are global over the [N=8, 4096] block (reference quirk)
__global__ void perk_kernel(const float* __restrict__ stats,
                            const float* __restrict__ gam,
                            const float* __restrict__ rgam,
                            float* __restrict__ feat) {
    int t = blockIdx.x * blockDim.x + threadIdx.x;
    if (t >= 64 * 3 * NB) return;
    int n = t & 7;
    int c = (t >> 3) % 3;
    int i = t / 24;
    float nneg = 0, npos = 0;
    for (int nn = 0; nn < NB; ++nn) {
        const float* s = stats + ((size_t)((nn * 3 + c) * 64) + i) * 6;
        nneg += s[0];
        npos += s[1];
    }
    const float* s = stats + ((size_t)((n * 3 + c) * 64) + i) * 6;
    float o3[3];
    aggd_finish(nneg, npos, s[2], s[3], s[4], s[5], 1.0f / 4096.0f, gam, rgam, o3);
    float* f = feat + (size_t)n * NFEAT + 9 + (i * 3 + c) * 3;
    f[0] = o3[0]; f[1] = o3[1]; f[2] = o3[2];
}

// base AGGD per channel: counts global over [8, 64*4096]
__global__ void base_kernel(const float* __restrict__ stats,
                            const float* __restrict__ gam,
                            const float* __restrict__ rgam,
                            float* __restrict__ feat) {
    int t = blockIdx.x * blockDim.x + threadIdx.x;
    if (t >= NB * 3) return;
    int n = t / 3, c = t % 3;
    float nneg = 0, npos = 0;
    for (int nn = 0; nn < NB; ++nn)
        for (int i = 0; i < 64; ++i) {
            const float* s = stats + ((size_t)((nn * 3 + c) * 64) + i) * 6;
            nneg += s[0];
            npos += s[1];
        }
    float sn = 0, sp = 0, sa = 0, ss = 0;
    for (int i = 0; i < 64; ++i) {
        const float* s = stats + ((size_t)((n * 3 + c) * 64) + i) * 6;
        sn += s[2]; sp += s[3]; sa += s[4]; ss += s[5];
    }
    float o3[3];
    aggd_finish(nneg, npos, sn, sp, sa, ss, 1.0f / 262144.0f, gam, rgam, o3);
    float* f = feat + (size_t)n * NFEAT + c * 3;
    f[0] = o3[0]; f[1] = o3[1]; f[2] = o3[2];
}

// -------------------------------------------------------------- final loss
__global__ __launch_bounds__(256) void loss_kernel(const float* __restrict__ feat,
                                                   float* __restrict__ out) {
    __shared__ float sd[256];
    const int T = NB * NFEAT;          // 4680
    int tid = threadIdx.x;
    const float* fsl = feat;
    const float* fsr = feat + T;
    const float* fhl = feat + 2 * T;
    const float* fhr = feat + 3 * T;
    float acc = 0.f;
    for (int i = tid; i < T; i += 256) {
        float dl = fsl[i] - fhl[i];
        float dr = fsr[i] - fhr[i];
        acc += dl * dl + dr * dr;
    }
    sd[tid] = acc;
    __syncthreads();
    for (int off = 128; off > 0; off >>= 1) {
        if (tid < off) sd[tid] += sd[tid + off];
        __syncthreads();
    }
    if (tid == 0) out[0] = sd[0] / (float)T;
}

extern "C" void kernel_launch(void* const* d_in, const int* in_sizes, int n_in,
                              void* d_out, int out_size, void* d_ws, size_t ws_size,
                              hipStream_t stream) {
    (void)in_sizes; (void)n_in; (void)out_size; (void)ws_size;
    const float* pred = (const float*)d_in[0];     // [8,6,512,512]
    const float* target = (const float*)d_in[1];   // [8,6,512,512]
    const float* kern = (const float*)d_in[2];     // [64,64,3]
    float* out = (float*)d_out;
    float* ws = (float*)d_ws;

    size_t off = 0;
    float* gam   = ws + off; off += 9804;                 // Γ arg table
    float* rgam  = ws + off; off += 9804;                 // Γ-ratio table
    float* feat  = ws + off; off += (size_t)4 * NB * NFEAT;
    float* stats = ws + off; off += (size_t)1536 * 6;
    float* kpack = ws + off; off += (size_t)3 * 64 * 64;  // repacked KLT kernel
    float* mscn  = ws + off; off += (size_t)NB * 3 * HH * WW;
    float* klt   = ws + off; off += (size_t)NB * 3 * 64 * NPATCH;

    gamma_table_kernel<<<(NGAM + 255) / 256, 256, 0, stream>>>(gam, rgam);
    repack_kernel<<<48, 256, 0, stream>>>(kern, kpack);

    for (int t = 0; t < 4; ++t) {
        const float* img = (t < 2) ? pred : target;
        int ch_off = (t & 1) ? 3 : 0;
        mscn_kernel<<<24 * 256, 256, 0, stream>>>(img, ch_off, mscn);
        klt_wmma_kernel<<<768, 256, 0, stream>>>(mscn, kpack, klt);
        rowstats_kernel<<<1536, 256, 0, stream>>>(klt, stats);
        perk_kernel<<<6, 256, 0, stream>>>(stats, gam, rgam, feat + (size_t)t * NB * NFEAT);
        base_kernel<<<1, 32, 0, stream>>>(stats, gam, rgam, feat + (size_t)t * NB * NFEAT);
    }
    loss_kernel<<<1, 256, 0, stream>>>(feat, out);
}